// SGC_76922864272070
// MI455X (gfx1250) — compile-verified
//
#include <hip/hip_runtime.h>
#include <hip/hip_bf16.h>
#include <math.h>

typedef __attribute__((ext_vector_type(2))) float v2f;
typedef __attribute__((ext_vector_type(8))) float v8f;

#define NODES_PER_BLOCK 8   // waves per block for gemm / softmax

// ---------------------------------------------------------------------------
// Dense GEMM  O[n_rows, NOUT] = H[n_rows, K] @ W[K, NOUT]
// using V_WMMA_F32_16X16X4_F32 (fp32 exact). One wave per 16-row strip,
// NT = ceil(NOUT/16) column tiles held in accumulators.
// A frag (16x4):  lane&15 = M row, (lane>>4)*2 = K offset, 2 VGPRs (K,K+1)
// B frag (4x16):  lane&15 = N col, (lane>>4)*2 = K row,    2 VGPRs (K,K+1)
// C/D  (16x16):   lane&15 = N col, VGPR v -> M = v + (lane>>4)*8
// ---------------------------------------------------------------------------
template <int K, int NOUT, int NT>
__global__ __launch_bounds__(256) void gemm_wmma_f32(
    const float* __restrict__ H, const float* __restrict__ W,
    float* __restrict__ O, int n_rows) {
  const int lane  = threadIdx.x & 31;
  const int wave  = threadIdx.x >> 5;
  const int mtile = blockIdx.x * (blockDim.x >> 5) + wave;
  const int mtiles = (n_rows + 15) >> 4;
  if (mtile >= mtiles) return;                 // wave-uniform: EXEC stays full

  const int row  = mtile * 16 + (lane & 15);   // A-fragment row
  const int koff = (lane >> 4) * 2;            // which K pair this half-wave holds
  const int col0 = lane & 15;                  // B/C/D column within tile

  v8f acc[NT];
#pragma unroll
  for (int t = 0; t < NT; ++t) acc[t] = (v8f){0.f,0.f,0.f,0.f,0.f,0.f,0.f,0.f};

  const float* hrow = H + (size_t)row * K;
#pragma unroll 4
  for (int k0 = 0; k0 < K; k0 += 4) {
    v2f a;
    a.x = hrow[k0 + koff];
    a.y = hrow[k0 + koff + 1];
#pragma unroll
    for (int t = 0; t < NT; ++t) {
      const int col = t * 16 + col0;
      v2f b;
      if (col < NOUT) {
        b.x = W[(size_t)(k0 + koff)     * NOUT + col];
        b.y = W[(size_t)(k0 + koff + 1) * NOUT + col];
      } else {
        b.x = 0.f; b.y = 0.f;
      }
      acc[t] = __builtin_amdgcn_wmma_f32_16x16x4_f32(
          /*neg_a=*/false, a, /*neg_b=*/false, b,
          /*c_mod=*/(short)0, acc[t], /*reuse_a=*/false, /*reuse_b=*/false);
    }
  }

  const int rbase = mtile * 16 + (lane >> 4) * 8;
#pragma unroll
  for (int t = 0; t < NT; ++t) {
    const int col = t * 16 + col0;
    if (col < NOUT) {
#pragma unroll
      for (int v = 0; v < 8; ++v) {
        const int r = rbase + v;
        if (r < n_rows) O[(size_t)r * NOUT + col] = acc[t][v];
      }
    }
  }
}

// ---------------------------------------------------------------------------
// Zero-fill (graph-capture safe, run before each scatter accumulation)
// ---------------------------------------------------------------------------
__global__ __launch_bounds__(256) void zero_f32(float* __restrict__ p, size_t n) {
  size_t i = (size_t)blockIdx.x * blockDim.x + threadIdx.x;
  if (i < n) p[i] = 0.f;
}

// ---------------------------------------------------------------------------
// Edge scatter-add: out[dst[e], :] += hw[src[e], :]
// DV4 = D/4 threads per edge, each moves a float4 (L2-resident traffic).
// ---------------------------------------------------------------------------
template <int DV4>
__global__ __launch_bounds__(256) void scatter_add(
    const float* __restrict__ hw, const int* __restrict__ src,
    const int* __restrict__ dst, float* __restrict__ out, int n_edges) {
  const int D = DV4 * 4;
  const unsigned idx = blockIdx.x * blockDim.x + threadIdx.x;
  const unsigned e = idx / (unsigned)DV4;
  if (e >= (unsigned)n_edges) return;
  const unsigned c = (idx % (unsigned)DV4) * 4;
  const int s = src[e];
  const int t = dst[e];
  const float4 v = *reinterpret_cast<const float4*>(hw + (size_t)s * D + c);
  float* o = out + (size_t)t * D + c;
  atomicAdd(o + 0, v.x);
  atomicAdd(o + 1, v.y);
  atomicAdd(o + 2, v.z);
  atomicAdd(o + 3, v.w);
}

// ---------------------------------------------------------------------------
// In-place log_softmax over 40 features; one wave32 per node.
// Lane l covers features l and l+32 (second valid only for l < 8).
// ---------------------------------------------------------------------------
__global__ __launch_bounds__(256) void log_softmax40(float* __restrict__ out,
                                                     int n_rows) {
  const int lane = threadIdx.x & 31;
  const int wave = threadIdx.x >> 5;
  const int row  = blockIdx.x * (blockDim.x >> 5) + wave;
  if (row >= n_rows) return;
  float* p = out + (size_t)row * 40;

  const float v0 = p[lane];
  const float v1 = (lane < 8) ? p[lane + 32] : -INFINITY;

  float m = fmaxf(v0, v1);
#pragma unroll
  for (int o = 16; o > 0; o >>= 1) m = fmaxf(m, __shfl_xor(m, o, 32));

  float s = __expf(v0 - m) + ((lane < 8) ? __expf(v1 - m) : 0.f);
#pragma unroll
  for (int o = 16; o > 0; o >>= 1) s += __shfl_xor(s, o, 32);

  const float lse = m + __logf(s);
  p[lane] = v0 - lse;
  if (lane < 8) p[lane + 32] = v1 - lse;
}

// ---------------------------------------------------------------------------
// Orchestration
// ---------------------------------------------------------------------------
extern "C" void kernel_launch(void* const* d_in, const int* in_sizes, int n_in,
                              void* d_out, int out_size, void* d_ws, size_t ws_size,
                              hipStream_t stream) {
  const float* x    = (const float*)d_in[0];   // [N,128]
  const int*   esrc = (const int*)  d_in[1];   // [E]
  const int*   edst = (const int*)  d_in[2];   // [E]
  const float* W1   = (const float*)d_in[3];   // [128,64]
  const float* W2   = (const float*)d_in[4];   // [64,64]
  const float* W3   = (const float*)d_in[5];   // [64,40]
  float* out = (float*)d_out;                  // [N,40]

  const int N = in_sizes[0] / 128;
  const int E = in_sizes[1];

  float* bufA = (float*)d_ws;                  // N*64
  float* bufB = bufA + (size_t)N * 64;         // N*64
  float* bufC = bufB + (size_t)N * 64;         // N*64

  const int mtiles = (N + 15) >> 4;
  const int gemm_blocks = (mtiles + NODES_PER_BLOCK - 1) / NODES_PER_BLOCK;
  const size_t n64 = (size_t)N * 64;
  const int zero_blocks64 = (int)((n64 + 255) / 256);
  const int sc64_blocks = (int)(((size_t)E * 16 + 255) / 256);
  const int sc40_blocks = (int)(((size_t)E * 10 + 255) / 256);
  const size_t n40 = (size_t)N * 40;
  const int zero_blocks40 = (int)((n40 + 255) / 256);
  const int sm_blocks = (N + NODES_PER_BLOCK - 1) / NODES_PER_BLOCK;

  // Layer 1: hw1 = x @ W1 -> bufA ; h1 = scatter(hw1) -> bufB
  gemm_wmma_f32<128, 64, 4><<<gemm_blocks, 256, 0, stream>>>(x, W1, bufA, N);
  zero_f32<<<zero_blocks64, 256, 0, stream>>>(bufB, n64);
  scatter_add<16><<<sc64_blocks, 256, 0, stream>>>(bufA, esrc, edst, bufB, E);

  // Layer 2: hw2 = h1 @ W2 -> bufC ; h2 = scatter(hw2) -> bufA
  gemm_wmma_f32<64, 64, 4><<<gemm_blocks, 256, 0, stream>>>(bufB, W2, bufC, N);
  zero_f32<<<zero_blocks64, 256, 0, stream>>>(bufA, n64);
  scatter_add<16><<<sc64_blocks, 256, 0, stream>>>(bufC, esrc, edst, bufA, E);

  // Layer 3: hw3 = h2 @ W3 -> bufB ; h3 = scatter(hw3) -> d_out
  gemm_wmma_f32<64, 40, 3><<<gemm_blocks, 256, 0, stream>>>(bufA, W3, bufB, N);
  zero_f32<<<zero_blocks40, 256, 0, stream>>>(out, n40);
  scatter_add<10><<<sc40_blocks, 256, 0, stream>>>(bufB, esrc, edst, out, E);

  // log_softmax in place on d_out
  log_softmax40<<<sm_blocks, 256, 0, stream>>>(out, N);
}